// BiLSTM_CRF_4715874091174
// MI455X (gfx1250) — compile-verified
//
#include <hip/hip_runtime.h>
#include <math.h>

// ---------------------------------------------------------------------------
// BiLSTM-CRF forward score for MI455X (gfx1250, wave32).
//   V=100000, E=256, H2=256, G=4*H2=1024, T=6, L=2048
// Pipeline:
//   K1  input_proj_wmma : pre = embed[sentence] @ Wih.T + bih + bhh
//                         fp32 WMMA 16x16x4; B panel double-buffered in LDS
//                         via global_load_async_to_lds_b64 (ASYNCcnt path)
//   K2  lstm_recurrent  : sequential scan, 1 WGP per direction; pre[t] loads
//                         software-pipelined one step ahead of the recurrence
//   K3  output_proj     : feats = [hf|hb] @ W_out.T + b_out
//   K4  crf_score       : CRF forward log-partition + gold path, scalar scan
// ---------------------------------------------------------------------------

typedef __attribute__((ext_vector_type(2))) float v2f;
typedef __attribute__((ext_vector_type(8))) float v8f;

#define L_SEQ   2048
#define E_DIM   256
#define H2_DIM  256
#define G_DIM   1024   // 4*H2
#define T_TAGS  6
#define START_TAG 4
#define STOP_TAG  5
#define NEG_INF (-10000.0f)

#define KCHUNK  32            // K elements staged per LDS chunk
#define NCHUNKS (E_DIM / KCHUNK)
// LDS B panel layout per buffer: [q(0..7)][half(0..1)][n(0..63)][pair(0..1)]
// float index = ((q*2+half)*64 + n)*2 + pair  -> one aligned b64 per cell.
#define BSH_FLOATS (8 * 2 * 64 * 2)   // 2048 floats = 8KB per buffer

__device__ __forceinline__ void async_copy_b64_to_lds(unsigned lds_off,
                                                      const void* gsrc) {
  unsigned long long ga = (unsigned long long)gsrc;
  asm volatile("global_load_async_to_lds_b64 %0, %1, off"
               :: "v"(lds_off), "v"(ga) : "memory");
}
__device__ __forceinline__ void wait_asynccnt0() {
  asm volatile("s_wait_asynccnt 0x0" ::: "memory");
}

// ---------------------------------------------------------------------------
// K1: pre[l, n] = sum_k embed[sentence[l], k] * Wih[n, k] + bih[n] + bhh[n]
// Block = 256 threads (8 waves). All 8 waves share one 64-wide N-group whose
// B panel is staged in LDS; each wave owns one 16-row M tile.
// Grid: (G_DIM/64, L_SEQ/128) = (16, 16).
// WMMA f32 16x16x4 operand layout (ISA 7.12.2):
//   A: lane<16 -> {K+0,K+1} of row M=lane&15 ; lane>=16 -> {K+2,K+3}
//   B: same K split, N = lane&15
//   C/D v8f: element (r,lane) -> M = r + 8*(lane>=16), N = lane&15
// ---------------------------------------------------------------------------
__global__ __launch_bounds__(256)
void input_proj_wmma(const int* __restrict__ sentence,
                     const float* __restrict__ embed,
                     const float* __restrict__ Wih,   // [G_DIM, E_DIM]
                     const float* __restrict__ bih,
                     const float* __restrict__ bhh,
                     float* __restrict__ pre) {       // [L_SEQ, G_DIM]
  __shared__ float Bsh[2][BSH_FLOATS];

  const int ti    = threadIdx.x;          // 0..255
  const int wave  = ti >> 5;              // 0..7  -> M tile within block
  const int lane  = ti & 31;
  const int mr    = lane & 15;
  const int hl    = (lane >> 4) & 1;      // lane half selects K pair
  const int ko    = hl * 2;
  const int nBase = blockIdx.x * 64;
  const int mTile = blockIdx.y * 128 + wave * 16;

  const long xrow = (long)sentence[mTile + mr] * E_DIM;

  // --- stage chunk 0 into buffer 0 ------------------------------------------
  {
#pragma unroll
    for (int it = 0; it < 4; ++it) {
      const int cell = it * 256 + ti;          // 1024 b64 cells per chunk
      const int n = cell & 63;
      const int h = (cell >> 6) & 1;
      const int q = cell >> 7;
      const float* src = Wih + (long)(nBase + n) * E_DIM + q * 4 + h * 2;
      const unsigned dst =
          (unsigned)(uintptr_t)&Bsh[0][((q * 2 + h) * 64 + n) * 2];
      async_copy_b64_to_lds(dst, src);
    }
  }

  v8f acc[4] = {};

  for (int c = 0; c < NCHUNKS; ++c) {
    const int buf = c & 1;
    wait_asynccnt0();          // chunk c resident (async loads complete in order)
    __syncthreads();           // all staging writes visible block-wide

    if (c + 1 < NCHUNKS) {     // stage next chunk into the other buffer
      const int cb = (c + 1) * KCHUNK;
#pragma unroll
      for (int it = 0; it < 4; ++it) {
        const int cell = it * 256 + ti;
        const int n = cell & 63;
        const int h = (cell >> 6) & 1;
        const int q = cell >> 7;
        const float* src = Wih + (long)(nBase + n) * E_DIM + cb + q * 4 + h * 2;
        const unsigned dst =
            (unsigned)(uintptr_t)&Bsh[buf ^ 1][((q * 2 + h) * 64 + n) * 2];
        async_copy_b64_to_lds(dst, src);
      }
    }

    const int cbase = c * KCHUNK;
#pragma unroll
    for (int q = 0; q < KCHUNK / 4; ++q) {
      v2f a = *(const v2f*)(embed + xrow + cbase + q * 4 + ko);
      const int bb = (q * 2 + hl) * 64;
      v2f b0 = *(const v2f*)&Bsh[buf][(bb +  0 + mr) * 2];
      v2f b1 = *(const v2f*)&Bsh[buf][(bb + 16 + mr) * 2];
      v2f b2 = *(const v2f*)&Bsh[buf][(bb + 32 + mr) * 2];
      v2f b3 = *(const v2f*)&Bsh[buf][(bb + 48 + mr) * 2];
      acc[0] = __builtin_amdgcn_wmma_f32_16x16x4_f32(false, a, false, b0,
                                                     (short)0, acc[0], false, false);
      acc[1] = __builtin_amdgcn_wmma_f32_16x16x4_f32(false, a, false, b1,
                                                     (short)0, acc[1], false, false);
      acc[2] = __builtin_amdgcn_wmma_f32_16x16x4_f32(false, a, false, b2,
                                                     (short)0, acc[2], false, false);
      acc[3] = __builtin_amdgcn_wmma_f32_16x16x4_f32(false, a, false, b3,
                                                     (short)0, acc[3], false, false);
    }
    __syncthreads();           // everyone done reading buf before it is re-staged
  }

  const int rowOff = hl * 8;
#pragma unroll
  for (int j = 0; j < 4; ++j) {
    const int col  = nBase + j * 16 + mr;
    const float bb = bih[col] + bhh[col];
#pragma unroll
    for (int r = 0; r < 8; ++r) {
      const int row = mTile + rowOff + r;
      pre[(long)row * G_DIM + col] = acc[j][r] + bb;
    }
  }
}

// ---------------------------------------------------------------------------
// K2: sequential LSTM scan. Grid = 2 blocks (dir), block = 1024 threads.
// Thread tid owns gate-row tid of z = pre[t] + Whh @ h.  h broadcast via LDS,
// Whh rows streamed as float4 (1 MB/direction, resident in the 192 MB L2).
// pre[t] is loaded one step ahead into a register so its L2 latency hides
// behind the 64-iteration dot product on the serial critical path.
// ---------------------------------------------------------------------------
__global__ __launch_bounds__(1024)
void lstm_recurrent(const float* __restrict__ pre_f,
                    const float* __restrict__ pre_b,
                    const float* __restrict__ Whh_f,   // [G_DIM, H2_DIM]
                    const float* __restrict__ Whh_b,
                    const float* __restrict__ h0,      // [2, H2]
                    const float* __restrict__ c0,      // [2, H2]
                    float* __restrict__ hf,            // [L_SEQ, H2]
                    float* __restrict__ hb) {          // [L_SEQ, H2]
  const int dir = blockIdx.x;
  const int tid = threadIdx.x;

  const float* __restrict__ pre = dir ? pre_b : pre_f;
  const float* __restrict__ Whh = dir ? Whh_b : Whh_f;
  float* __restrict__ hs        = dir ? hb : hf;

  __shared__ float h_sh[H2_DIM];
  __shared__ float z_sh[G_DIM];

  float c_reg = 0.0f;
  if (tid < H2_DIM) {
    h_sh[tid] = h0[dir * H2_DIM + tid];
    c_reg     = c0[dir * H2_DIM + tid];
  }
  __syncthreads();

  const float4* __restrict__ wrow =
      reinterpret_cast<const float4*>(Whh + (long)tid * H2_DIM);

  float zin = pre[(long)(dir ? (L_SEQ - 1) : 0) * G_DIM + tid];

  for (int s = 0; s < L_SEQ; ++s) {
    const int t = dir ? (L_SEQ - 1 - s) : s;
    // issue next step's pre load before the dot product (latency hiding)
    float zin_next = 0.0f;
    if (s + 1 < L_SEQ) {
      const int tn = dir ? (t - 1) : (t + 1);
      zin_next = pre[(long)tn * G_DIM + tid];
    }
    float z = zin;
#pragma unroll 8
    for (int q = 0; q < H2_DIM / 4; ++q) {
      const float4 w = wrow[q];
      z += w.x * h_sh[4 * q + 0] + w.y * h_sh[4 * q + 1]
         + w.z * h_sh[4 * q + 2] + w.w * h_sh[4 * q + 3];
    }
    z_sh[tid] = z;
    __syncthreads();
    if (tid < H2_DIM) {
      const float ig = 1.0f / (1.0f + __expf(-z_sh[tid]));
      const float fg = 1.0f / (1.0f + __expf(-z_sh[H2_DIM + tid]));
      const float gg = tanhf(z_sh[2 * H2_DIM + tid]);
      const float og = 1.0f / (1.0f + __expf(-z_sh[3 * H2_DIM + tid]));
      c_reg = fg * c_reg + ig * gg;
      const float h = og * tanhf(c_reg);
      h_sh[tid] = h;
      hs[(long)t * H2_DIM + tid] = h;
    }
    __syncthreads();
    zin = zin_next;
  }
}

// ---------------------------------------------------------------------------
// K3: feats[l, t] = b_out[t] + hf[l]·W_out[t, :256] + hb[l]·W_out[t, 256:]
// ---------------------------------------------------------------------------
__global__ __launch_bounds__(256)
void output_proj(const float* __restrict__ hf,
                 const float* __restrict__ hb,
                 const float* __restrict__ W_out,   // [T, 2*H2]
                 const float* __restrict__ b_out,   // [T]
                 float* __restrict__ feats) {       // [L, T]
  const int idx = blockIdx.x * blockDim.x + threadIdx.x;
  if (idx >= L_SEQ * T_TAGS) return;
  const int l = idx / T_TAGS;
  const int t = idx - l * T_TAGS;
  const float* __restrict__ wf  = W_out + (long)t * (2 * H2_DIM);
  const float* __restrict__ wb  = wf + H2_DIM;
  const float* __restrict__ hfl = hf + (long)l * H2_DIM;
  const float* __restrict__ hbl = hb + (long)l * H2_DIM;
  float s = b_out[t];
#pragma unroll 4
  for (int k = 0; k < H2_DIM; ++k)
    s += hfl[k] * wf[k] + hbl[k] * wb[k];
  feats[idx] = s;
}

// ---------------------------------------------------------------------------
// K4: CRF forward log-partition + gold path score; sequential scalar scan
// (T=6, ~150K flops total; one lane is fine).
// ---------------------------------------------------------------------------
__global__ void crf_score(const float* __restrict__ feats,    // [L, T]
                          const float* __restrict__ trans,    // [T, T]
                          const int* __restrict__ tags,       // [L]
                          float* __restrict__ out) {
  if (threadIdx.x != 0 || blockIdx.x != 0) return;

  float tr[T_TAGS * T_TAGS];
  for (int i = 0; i < T_TAGS * T_TAGS; ++i) tr[i] = trans[i];

  float fv[T_TAGS];
  for (int t = 0; t < T_TAGS; ++t) fv[t] = (t == START_TAG) ? 0.0f : NEG_INF;

  for (int l = 0; l < L_SEQ; ++l) {
    float fvn[T_TAGS];
    for (int nt = 0; nt < T_TAGS; ++nt) {
      float m = -3.0e38f;
      for (int p = 0; p < T_TAGS; ++p)
        m = fmaxf(m, fv[p] + tr[nt * T_TAGS + p]);
      float ssum = 0.0f;
      for (int p = 0; p < T_TAGS; ++p)
        ssum += __expf(fv[p] + tr[nt * T_TAGS + p] - m);
      fvn[nt] = m + __logf(ssum) + feats[l * T_TAGS + nt];
    }
    for (int t = 0; t < T_TAGS; ++t) fv[t] = fvn[t];
  }

  float m = -3.0e38f;
  for (int t = 0; t < T_TAGS; ++t)
    m = fmaxf(m, fv[t] + tr[STOP_TAG * T_TAGS + t]);
  float ssum = 0.0f;
  for (int t = 0; t < T_TAGS; ++t)
    ssum += __expf(fv[t] + tr[STOP_TAG * T_TAGS + t] - m);
  const float forward_score = m + __logf(ssum);

  float gold = 0.0f;
  int prev = START_TAG;
  for (int l = 0; l < L_SEQ; ++l) {
    const int cur = tags[l];
    gold += tr[cur * T_TAGS + prev] + feats[l * T_TAGS + cur];
    prev = cur;
  }
  gold += tr[STOP_TAG * T_TAGS + prev];

  out[0] = forward_score - gold;
}

// ---------------------------------------------------------------------------
extern "C" void kernel_launch(void* const* d_in, const int* in_sizes, int n_in,
                              void* d_out, int out_size, void* d_ws, size_t ws_size,
                              hipStream_t stream) {
  (void)in_sizes; (void)n_in; (void)out_size; (void)ws_size;

  const int*   sentence = (const int*)  d_in[0];
  const int*   tags     = (const int*)  d_in[1];
  const float* embed    = (const float*)d_in[2];
  const float* Wih_f    = (const float*)d_in[3];
  const float* Whh_f    = (const float*)d_in[4];
  const float* bih_f    = (const float*)d_in[5];
  const float* bhh_f    = (const float*)d_in[6];
  const float* Wih_b    = (const float*)d_in[7];
  const float* Whh_b    = (const float*)d_in[8];
  const float* bih_b    = (const float*)d_in[9];
  const float* bhh_b    = (const float*)d_in[10];
  const float* h0       = (const float*)d_in[11];
  const float* c0       = (const float*)d_in[12];
  const float* W_out    = (const float*)d_in[13];
  const float* b_out    = (const float*)d_in[14];
  const float* trans    = (const float*)d_in[15];

  float* ws    = (float*)d_ws;
  float* pre_f = ws;                       ws += (long)L_SEQ * G_DIM;
  float* pre_b = ws;                       ws += (long)L_SEQ * G_DIM;
  float* hf    = ws;                       ws += (long)L_SEQ * H2_DIM;
  float* hb    = ws;                       ws += (long)L_SEQ * H2_DIM;
  float* feats = ws;                       ws += (long)L_SEQ * T_TAGS;

  dim3 g1(G_DIM / 64, L_SEQ / 128);  // (16, 16)
  dim3 b1(256);
  input_proj_wmma<<<g1, b1, 0, stream>>>(sentence, embed, Wih_f, bih_f, bhh_f, pre_f);
  input_proj_wmma<<<g1, b1, 0, stream>>>(sentence, embed, Wih_b, bih_b, bhh_b, pre_b);

  lstm_recurrent<<<2, 1024, 0, stream>>>(pre_f, pre_b, Whh_f, Whh_b, h0, c0, hf, hb);

  output_proj<<<(L_SEQ * T_TAGS + 255) / 256, 256, 0, stream>>>(hf, hb, W_out, b_out, feats);

  crf_score<<<1, 32, 0, stream>>>(feats, trans, tags, (float*)d_out);
}